// GCN_regression_87282325390048
// MI455X (gfx1250) — compile-verified
//
#include <hip/hip_runtime.h>
#include <math.h>

#define N_NODES 100000
#define E_EDGES 1600000
#define LRELU_ALPHA 0.2f

typedef float v2f __attribute__((ext_vector_type(2)));
typedef float v8f __attribute__((ext_vector_type(8)));

// ---------------------------------------------------------------------------
// small helper kernels
// ---------------------------------------------------------------------------
__global__ void zero_f_kernel(float* __restrict__ p, int n) {
    int i = blockIdx.x * blockDim.x + threadIdx.x;
    if (i < n) p[i] = 0.0f;
}

__global__ void deg_count_kernel(const int* __restrict__ dst,
                                 float* __restrict__ deg, int e) {
    int i = blockIdx.x * blockDim.x + threadIdx.x;
    if (i < e) atomicAdd(&deg[dst[i]], 1.0f);
}

__global__ void dis_kernel(const float* __restrict__ deg,
                           float* __restrict__ dis, int n) {
    int i = blockIdx.x * blockDim.x + threadIdx.x;
    if (i < n) dis[i] = rsqrtf(deg[i] + 1.0f);  // +1 for the self loop
}

// ---------------------------------------------------------------------------
// WMMA fp32 GEMM:  out[M,NOUT] = act(A[M,K] @ W[K,NOUT] (+ bias))
// One wave computes a 16 x NOUT strip.
// W is staged in LDS in PAIR-MAJOR layout:
//   sW[(kk/2)*NOUT*2 + col*2 + (kk&1)] = W[kk][col]
// so a lane's B fragment {W[kk][col], W[kk+1][col]} is one aligned ds_load_b64.
// ACT: 0 = none, 1 = leaky_relu(0.2), 2 = relu
// ---------------------------------------------------------------------------
template <int K, int NOUT, int ACT, bool HAS_BIAS>
__global__ __launch_bounds__(256)
void gemm_wmma_f32_kernel(const float* __restrict__ A,
                          const float* __restrict__ W,
                          const float* __restrict__ bias,
                          float* __restrict__ out, int M) {
    constexpr int NT = NOUT / 16;               // number of 16-wide col tiles
    __shared__ float sW[K * NOUT];              // pair-major swizzled copy of W

    const int tid = threadIdx.x;
    for (int i = tid; i < K * NOUT; i += 256) {
        const int kk  = i / NOUT;
        const int col = i - kk * NOUT;
        sW[((kk >> 1) * NOUT + col) * 2 + (kk & 1)] = W[i];
    }
    __syncthreads();

    const int wave  = tid >> 5;
    const int lane  = tid & 31;
    const int lhalf = lane >> 4;                // 0 | 1
    const int lmod  = lane & 15;

    const int row0 = (blockIdx.x * 8 + wave) * 16;
    int rowa = row0 + lmod;                     // A-fragment row for this lane
    rowa = rowa < M ? rowa : (M - 1);           // clamp (no EXEC divergence)

    v8f acc[NT] = {};

#pragma unroll
    for (int k = 0; k < K; k += 4) {
        // A fragment: lanes 0-15 hold K=k,k+1; lanes 16-31 hold K=k+2,k+3
        v2f a = *(const v2f*)(A + (size_t)rowa * K + k + 2 * lhalf);
        const int p = (k >> 1) + lhalf;         // W pair index for this lane
#pragma unroll
        for (int t = 0; t < NT; ++t) {
            // one aligned 8-byte LDS load -> {W[kk][col], W[kk+1][col]}
            v2f b = *(const v2f*)(sW + 2 * (p * NOUT + t * 16 + lmod));
            acc[t] = __builtin_amdgcn_wmma_f32_16x16x4_f32(
                false, a, false, b, (short)0, acc[t], false, false);
        }
    }

    const int rowbase = row0 + 8 * lhalf;       // C layout: vgpr r -> row r (+8 hi half)
#pragma unroll
    for (int t = 0; t < NT; ++t) {
        const int col = t * 16 + lmod;
        float bv = HAS_BIAS ? bias[col] : 0.0f;
#pragma unroll
        for (int r = 0; r < 8; ++r) {
            const int row = rowbase + r;
            if (row < M) {
                float v = acc[t][r] + bv;
                if (ACT == 1) v = v > 0.0f ? v : LRELU_ALPHA * v;
                if (ACT == 2) v = v > 0.0f ? v : 0.0f;
                out[(size_t)row * NOUT + col] = v;
            }
        }
    }
}

// ---------------------------------------------------------------------------
// GCN aggregation: agg[v] = t[v]*dis[v]^2 + sum_{(s->v)} t[s]*dis[s]*dis[v]
// ---------------------------------------------------------------------------
template <int F>
__global__ void agg_init_kernel(const float* __restrict__ t,
                                const float* __restrict__ dis,
                                float* __restrict__ agg, int n) {
    int i = blockIdx.x * blockDim.x + threadIdx.x;
    if (i < n * F) {
        int v = i / F;
        float d = dis[v];
        agg[i] = t[i] * d * d;                  // self-loop term
    }
}

template <int F>
__global__ void agg_edge_kernel(const float* __restrict__ t,
                                const float* __restrict__ dis,
                                const int* __restrict__ src,
                                const int* __restrict__ dst,
                                float* __restrict__ agg, int e) {
    int i = blockIdx.x * blockDim.x + threadIdx.x;
    if (i < e * F) {
        int ed = i / F;
        int f  = i - ed * F;
        int s = src[ed];
        int d = dst[ed];
        float w = dis[s] * dis[d];
        atomicAdd(&agg[(size_t)d * F + f], t[(size_t)s * F + f] * w);
    }
}

template <int F>
__global__ void finalize_relu_kernel(float* __restrict__ agg,
                                     const float* __restrict__ bias, int n) {
    int i = blockIdx.x * blockDim.x + threadIdx.x;
    if (i < n * F) {
        float v = agg[i] + bias[i % F];
        agg[i] = v > 0.0f ? v : 0.0f;
    }
}

// ---------------------------------------------------------------------------
// final linear: out[v] = h2[v,:] @ W3 + b3     (K=32 -> plain VALU dot)
// ---------------------------------------------------------------------------
__global__ void final_linear_kernel(const float* __restrict__ h,
                                    const float* __restrict__ W3,
                                    const float* __restrict__ b3,
                                    float* __restrict__ out, int n) {
    int v = blockIdx.x * blockDim.x + threadIdx.x;
    if (v < n) {
        float acc = b3[0];
#pragma unroll
        for (int j = 0; j < 32; ++j) acc += h[(size_t)v * 32 + j] * W3[j];
        out[v] = acc;
    }
}

// ---------------------------------------------------------------------------
extern "C" void kernel_launch(void* const* d_in, const int* in_sizes, int n_in,
                              void* d_out, int out_size, void* d_ws, size_t ws_size,
                              hipStream_t stream) {
    (void)in_sizes; (void)n_in; (void)out_size; (void)ws_size;

    const float* x  = (const float*)d_in[0];
    /* d_in[1] = adj (unused by the reference forward) */
    const int*   ei = (const int*)d_in[2];
    const float* W0 = (const float*)d_in[3];
    const float* b0 = (const float*)d_in[4];
    const float* W1 = (const float*)d_in[5];
    const float* b1 = (const float*)d_in[6];
    const float* W2 = (const float*)d_in[7];
    const float* b2 = (const float*)d_in[8];
    const float* W3 = (const float*)d_in[9];
    const float* b3 = (const float*)d_in[10];
    float* out = (float*)d_out;

    const int n = N_NODES;
    const int e = E_EDGES;
    const int* src = ei;          // edge_index[0,:]
    const int* dst = ei + e;      // edge_index[1,:]

    // workspace layout (floats): dis[N] deg[N] h0[N*64] t[N*64] agg[N*64]
    float* ws  = (float*)d_ws;
    float* dis = ws;               ws += n;
    float* deg = ws;               ws += n;
    float* h0  = ws;               ws += (size_t)n * 64;
    float* t   = ws;               ws += (size_t)n * 64;
    float* agg = ws;               ws += (size_t)n * 64;

    const int B = 256;

    // --- degree normalization -------------------------------------------
    zero_f_kernel<<<(n + B - 1) / B, B, 0, stream>>>(deg, n);
    deg_count_kernel<<<(e + B - 1) / B, B, 0, stream>>>(dst, deg, e);
    dis_kernel<<<(n + B - 1) / B, B, 0, stream>>>(deg, dis, n);

    // 8 waves/block, 16 rows/wave => 128 rows per block
    const int gemm_blocks = (n + 127) / 128;

    // --- layer 0: h0 = leaky_relu(x @ W0 + b0) --------------------------
    gemm_wmma_f32_kernel<128, 64, 1, true>
        <<<gemm_blocks, B, 0, stream>>>(x, W0, b0, h0, n);

    // --- conv1: h1 = relu(scatter(h0 @ W1) + b1)   (h1 lives in agg) ----
    gemm_wmma_f32_kernel<64, 64, 0, false>
        <<<gemm_blocks, B, 0, stream>>>(h0, W1, nullptr, t, n);
    agg_init_kernel<64><<<((n * 64) + B - 1) / B, B, 0, stream>>>(t, dis, agg, n);
    agg_edge_kernel<64><<<((e * 64) + B - 1) / B, B, 0, stream>>>(t, dis, src, dst, agg, e);
    finalize_relu_kernel<64><<<((n * 64) + B - 1) / B, B, 0, stream>>>(agg, b1, n);

    // --- conv2: h2 = relu(scatter(h1 @ W2) + b2)   (h2 reuses h0 buffer) -
    gemm_wmma_f32_kernel<64, 32, 0, false>
        <<<gemm_blocks, B, 0, stream>>>(agg, W2, nullptr, t, n);
    agg_init_kernel<32><<<((n * 32) + B - 1) / B, B, 0, stream>>>(t, dis, h0, n);
    agg_edge_kernel<32><<<((e * 32) + B - 1) / B, B, 0, stream>>>(t, dis, src, dst, h0, e);
    finalize_relu_kernel<32><<<((n * 32) + B - 1) / B, B, 0, stream>>>(h0, b2, n);

    // --- output head ----------------------------------------------------
    final_linear_kernel<<<(n + B - 1) / B, B, 0, stream>>>(h0, W3, b3, out, n);
}